// GNN_72911364817162
// MI455X (gfx1250) — compile-verified
//
#include <hip/hip_runtime.h>

// ---------------------------------------------------------------------------
// GNN (typed linear + 2-layer GCN) for MI455X / gfx1250.
//
// - 4 propagation hops dominate (~1.6 GB L2-resident gather + f32 atomic
//   traffic per hop) -> wave-per-edge float4 gathers + no-return atomics.
// - Three 256x256 fp32 GEMMs on V_WMMA_F32_16X16X4_F32.  Nodes are
//   counting-sorted by type with buckets PADDED to multiples of 16, so every
//   WMMA tile is single-type: no masking, no EXEC manipulation, pad rows are
//   zeros in LDS.  Weights are repacked K-pair-interleaved so each B fragment
//   is one coalesced global_load_b64; A fragments are one ds_load_b64 from a
//   pitch-260 LDS tile (perfect 64-bank spread for wave32 b64 reads).
// ---------------------------------------------------------------------------

typedef __attribute__((ext_vector_type(2))) float v2f;
typedef __attribute__((ext_vector_type(8))) float v8f;

#define DIM       256
#define NUM_TYPES   8
#define LDA       260   // pitch: 260 mod 64 == 4 -> b64 reads hit all 64 banks once

// ---------------- utility kernels ----------------

__global__ void zero4_kernel(float4* __restrict__ p, long n4) {
  long i = (long)blockIdx.x * blockDim.x + threadIdx.x;
  long stride = (long)gridDim.x * blockDim.x;
  float4 z = make_float4(0.f, 0.f, 0.f, 0.f);
  for (; i < n4; i += stride) p[i] = z;
}

__global__ void fill_int_kernel(int* __restrict__ p, int v, int n) {
  int i = blockIdx.x * blockDim.x + threadIdx.x;
  if (i < n) p[i] = v;
}

__global__ void degree_kernel(const int* __restrict__ src, const int* __restrict__ dst,
                              float* __restrict__ deg_out, float* __restrict__ deg_in, int E) {
  int e = blockIdx.x * blockDim.x + threadIdx.x;
  if (e < E) {
    atomicAdd(&deg_out[src[e]], 1.0f);
    atomicAdd(&deg_in[dst[e]], 1.0f);
  }
}

__global__ void norm_kernel(const int* __restrict__ src, const int* __restrict__ dst,
                            const float* __restrict__ deg_out, const float* __restrict__ deg_in,
                            float* __restrict__ normv, int E) {
  int e = blockIdx.x * blockDim.x + threadIdx.x;
  if (e < E) {
    float a = __frsqrt_rn(fmaxf(deg_out[src[e]], 1.0f));
    float b = __frsqrt_rn(fmaxf(deg_in[dst[e]], 1.0f));
    normv[e] = a * b;
  }
}

// ---------------- type bucketing (counting sort, 16-padded buckets) --------

__global__ void bucket_init(int* __restrict__ counts, int* __restrict__ cursor) {
  if (threadIdx.x < NUM_TYPES) { counts[threadIdx.x] = 0; cursor[threadIdx.x] = 0; }
}

__global__ void bucket_count(const int* __restrict__ node_type, int* __restrict__ counts, int n) {
  int i = blockIdx.x * blockDim.x + threadIdx.x;
  if (i < n) atomicAdd(&counts[node_type[i]], 1);
}

// exclusive scan of counts, each bucket start rounded up to a 16 boundary
__global__ void bucket_prefix(const int* __restrict__ counts, int* __restrict__ offs) {
  if (blockIdx.x == 0 && threadIdx.x == 0) {
    int s = 0;
    for (int t = 0; t < NUM_TYPES; ++t) {
      offs[t] = s;
      s += ((counts[t] + 15) / 16) * 16;
    }
  }
}

__global__ void bucket_scatter(const int* __restrict__ node_type, const int* __restrict__ offs,
                               int* __restrict__ cursor, int* __restrict__ order, int n) {
  int i = blockIdx.x * blockDim.x + threadIdx.x;
  if (i < n) {
    int t = node_type[i];
    int pos = offs[t] + atomicAdd(&cursor[t], 1);
    order[pos] = i;
  }
}

// ---------------- weight repack: K-pair interleave -------------------------
// Wp[m][k2][n] = { W[m][2*k2][n], W[m][2*k2+1][n] }  -> B fragment = one b64.

__global__ void pack_w_kernel(const float* __restrict__ W, v2f* __restrict__ Wp, int total) {
  int i = blockIdx.x * blockDim.x + threadIdx.x;   // over mats*128*256
  if (i >= total) return;
  int n  = i & (DIM - 1);
  int k2 = (i >> 8) & 127;
  int m  = i >> 15;
  const float* Wm = W + (long)m * DIM * DIM;
  v2f p;
  p.x = Wm[(2 * k2) * DIM + n];
  p.y = Wm[(2 * k2 + 1) * DIM + n];
  Wp[i] = p;
}

// ---------------- WMMA tile core -------------------------------------------
// Block = 256 threads = 8 waves; tile = 16 rows x 256 cols.  Each wave owns
// two 16x16 col tiles sharing one A fragment (two independent WMMA chains).

__device__ __forceinline__ void wmma_tile_256(const float* As, const v2f* Bp,
                                              int wave, int lane, v8f& c0, v8f& c1) {
  int m16   = lane & 15;
  int krow  = lane >> 4;          // 0 or 1
  int khalf = krow * 2;           // 0 or 2
  int arow  = m16 * LDA;
  int bcol  = wave * 32 + m16;
#pragma unroll 8
  for (int k4 = 0; k4 < DIM / 4; ++k4) {
    v2f a  = *(const v2f*)&As[arow + 4 * k4 + khalf];
    const v2f* brow = Bp + (2 * k4 + krow) * DIM;
    v2f b0 = brow[bcol];
    v2f b1 = brow[bcol + 16];
    c0 = __builtin_amdgcn_wmma_f32_16x16x4_f32(false, a, false, b0, (short)0, c0, false, false);
    c1 = __builtin_amdgcn_wmma_f32_16x16x4_f32(false, a, false, b1, (short)0, c1, false, false);
  }
}

// ---------------- typed input linear (single-type padded tiles) ------------

__global__ __launch_bounds__(256) void input_gemm_wmma(
    const float* __restrict__ x, const int* __restrict__ order,
    const int* __restrict__ node_type, const v2f* __restrict__ fWp,
    const float* __restrict__ fb, float* __restrict__ out) {
  __shared__ float As[16 * LDA];
  __shared__ int nid[16];
  __shared__ int ttile_s;

  int tid = threadIdx.x;
  int base = blockIdx.x * 16;

  if (tid == 0) ttile_s = -1;
  __syncthreads();
  if (tid < 16) {
    int node = order[base + tid];
    nid[tid] = node;
    if (node >= 0) ttile_s = node_type[node];   // single type per padded tile
  }
  __syncthreads();
  int ttile = ttile_s;
  if (ttile < 0) return;   // fully padded tile (uniform exit)

  for (int r = 0; r < 16; ++r) {
    int node = nid[r];
    As[r * LDA + tid] = (node >= 0) ? x[(long)node * DIM + tid] : 0.0f;
  }
  __syncthreads();

  int wave = tid >> 5;
  int lane = tid & 31;
  v8f c0 = {}, c1 = {};
  wmma_tile_256(As, fWp + (long)ttile * (DIM / 2) * DIM, wave, lane, c0, c1);

  int m16 = lane & 15;
  int mbase = (lane >> 4) * 8;
  int ncol = wave * 32 + m16;
  float b0 = fb[ttile * DIM + ncol];
  float b1 = fb[ttile * DIM + ncol + 16];
#pragma unroll
  for (int v = 0; v < 8; ++v) {
    int node = nid[v + mbase];
    if (node >= 0) {
      out[(long)node * DIM + ncol]      = c0[v] + b0;
      out[(long)node * DIM + ncol + 16] = c1[v] + b1;
    }
  }
}

// ---------------- dense GCN layer linear (in-place safe) -------------------

__global__ __launch_bounds__(256) void layer_gemm_wmma(
    const float* __restrict__ in, const v2f* __restrict__ Wp,
    const float* __restrict__ bias, float* __restrict__ out, int n, int relu) {
  __shared__ float As[16 * LDA];

  int tid = threadIdx.x;
  int base = blockIdx.x * 16;

  for (int r = 0; r < 16; ++r) {
    int row = base + r;
    As[r * LDA + tid] = (row < n) ? in[(long)row * DIM + tid] : 0.0f;
  }
  __syncthreads();

  int wave = tid >> 5;
  int lane = tid & 31;
  v8f c0 = {}, c1 = {};
  wmma_tile_256(As, Wp, wave, lane, c0, c1);

  int m16 = lane & 15;
  int mbase = (lane >> 4) * 8;
  int ncol = wave * 32 + m16;
  float b0 = bias[ncol];
  float b1 = bias[ncol + 16];
#pragma unroll
  for (int v = 0; v < 8; ++v) {
    int row = base + v + mbase;
    if (row < n) {
      float v0 = c0[v] + b0;
      float v1 = c1[v] + b1;
      if (relu) { v0 = fmaxf(v0, 0.0f); v1 = fmaxf(v1, 0.0f); }
      out[(long)row * DIM + ncol]      = v0;
      out[(long)row * DIM + ncol + 16] = v1;
    }
  }
}

// ---------------- edge propagation: hout[dst] += h[src] * norm -------------
// One wave per edge: 2x float4 coalesced gather (512 B / wave / instr),
// 8 no-return f32 atomics per lane (global_atomic_add_f32, STOREcnt path).

__global__ __launch_bounds__(256) void edge_scatter(
    const float* __restrict__ h, const int* __restrict__ src,
    const int* __restrict__ dst, const float* __restrict__ normv,
    float* __restrict__ hout, int E) {
  int e = blockIdx.x * (blockDim.x >> 5) + (threadIdx.x >> 5);
  int lane = threadIdx.x & 31;
  if (e >= E) return;

  int s = src[e];
  int d = dst[e];
  float w = normv[e];

  const float4* hs = (const float4*)(h + (long)s * DIM);
  float4 v0 = hs[lane];
  float4 v1 = hs[lane + 32];

  float* ho = hout + (long)d * DIM;
  int c0 = lane * 4;
  atomicAdd(&ho[c0 + 0], v0.x * w);
  atomicAdd(&ho[c0 + 1], v0.y * w);
  atomicAdd(&ho[c0 + 2], v0.z * w);
  atomicAdd(&ho[c0 + 3], v0.w * w);
  atomicAdd(&ho[128 + c0 + 0], v1.x * w);
  atomicAdd(&ho[128 + c0 + 1], v1.y * w);
  atomicAdd(&ho[128 + c0 + 2], v1.z * w);
  atomicAdd(&ho[128 + c0 + 3], v1.w * w);
}

// ---------------- host-side launch ----------------

extern "C" void kernel_launch(void* const* d_in, const int* in_sizes, int n_in,
                              void* d_out, int out_size, void* d_ws, size_t ws_size,
                              hipStream_t stream) {
  const float* x     = (const float*)d_in[0];
  const int*   ntype = (const int*)d_in[1];
  const int*   eidx  = (const int*)d_in[2];
  const float* fW    = (const float*)d_in[3];
  const float* fb    = (const float*)d_in[4];
  const float* gW    = (const float*)d_in[5];
  const float* gb    = (const float*)d_in[6];

  int N = in_sizes[1];
  int E = in_sizes[2] / 2;
  const int* src = eidx;
  const int* dst = eidx + E;
  float* out = (float*)d_out;

  int tiles_pad = (N + 15) / 16 + NUM_TYPES;   // bucket padding upper bound
  int order_len = tiles_pad * 16;

  // workspace carve (keep 16B alignment)
  char* w = (char*)d_ws;
  size_t off = 0;
  float* buf0    = (float*)(w + off); off += (size_t)N * DIM * sizeof(float);
  v2f*   fWp     = (v2f*)  (w + off); off += (size_t)NUM_TYPES * (DIM / 2) * DIM * sizeof(v2f);
  v2f*   gWp     = (v2f*)  (w + off); off += (size_t)2 * (DIM / 2) * DIM * sizeof(v2f);
  float* deg_out = (float*)(w + off); off += (size_t)N * sizeof(float);
  float* deg_in  = (float*)(w + off); off += (size_t)N * sizeof(float);
  float* normv   = (float*)(w + off); off += (size_t)E * sizeof(float);
  int*   order   = (int*)  (w + off); off += ((size_t)order_len * sizeof(int) + 15) & ~(size_t)15;
  int*   counts  = (int*)  (w + off); off += 64;
  int*   cursor  = (int*)  (w + off); off += 64;
  int*   offs    = (int*)  (w + off); off += 64;

  int tiles = (N + 15) / 16;
  int eblk  = (E + 255) / 256;
  int nblk  = (N + 255) / 256;
  int sblk  = (E + 7) / 8;          // 8 edge-waves per 256-thread block
  long h4   = (long)N * DIM / 4;

  // degrees + edge norms (deg_out & deg_in contiguous -> one zero pass)
  zero4_kernel<<<256, 256, 0, stream>>>((float4*)deg_out, (2L * N) / 4);
  degree_kernel<<<eblk, 256, 0, stream>>>(src, dst, deg_out, deg_in, E);
  norm_kernel<<<eblk, 256, 0, stream>>>(src, dst, deg_out, deg_in, normv, E);

  // counting sort by type with 16-padded buckets (pad slots = -1)
  bucket_init<<<1, 32, 0, stream>>>(counts, cursor);
  bucket_count<<<nblk, 256, 0, stream>>>(ntype, counts, N);
  bucket_prefix<<<1, 1, 0, stream>>>(counts, offs);
  fill_int_kernel<<<(order_len + 255) / 256, 256, 0, stream>>>(order, -1, order_len);
  bucket_scatter<<<nblk, 256, 0, stream>>>(ntype, offs, cursor, order, N);

  // repack weights K-pair-interleaved for b64 B fragments
  int fwp_tot = NUM_TYPES * (DIM / 2) * DIM;
  int gwp_tot = 2 * (DIM / 2) * DIM;
  pack_w_kernel<<<(fwp_tot + 255) / 256, 256, 0, stream>>>(fW, fWp, fwp_tot);
  pack_w_kernel<<<(gwp_tot + 255) / 256, 256, 0, stream>>>(gW, gWp, gwp_tot);

  // typed input linear: x -> buf0 (padded tiles, uniform type per tile)
  input_gemm_wmma<<<tiles_pad, 256, 0, stream>>>(x, order, ntype, fWp, fb, buf0);

  // ---- layer 0: two hops, then linear + relu ----
  zero4_kernel<<<2048, 256, 0, stream>>>((float4*)out, h4);
  edge_scatter<<<sblk, 256, 0, stream>>>(buf0, src, dst, normv, out, E);
  zero4_kernel<<<2048, 256, 0, stream>>>((float4*)buf0, h4);
  edge_scatter<<<sblk, 256, 0, stream>>>(out, src, dst, normv, buf0, E);
  layer_gemm_wmma<<<tiles, 256, 0, stream>>>(buf0, gWp, gb, out, N, 1);

  // ---- layer 1: two hops, then linear (no relu), in-place into d_out ----
  zero4_kernel<<<2048, 256, 0, stream>>>((float4*)buf0, h4);
  edge_scatter<<<sblk, 256, 0, stream>>>(out, src, dst, normv, buf0, E);
  zero4_kernel<<<2048, 256, 0, stream>>>((float4*)out, h4);
  edge_scatter<<<sblk, 256, 0, stream>>>(buf0, src, dst, normv, out, E);
  layer_gemm_wmma<<<tiles, 256, 0, stream>>>(out, gWp + (DIM / 2) * DIM, gb + DIM, out, N, 0);
}